// LIF_75634374082823
// MI455X (gfx1250) — compile-verified
//
#include <hip/hip_runtime.h>

// LIF forward: T=64 sequential steps over B*N = 524288 independent neurons.
// Pure HBM streaming (~268 MB, floor ~11.5us @ 23.3 TB/s). No matmul is
// possible (reset nonlinearity), so the CDNA5 win is the async global->LDS
// DMA pipeline + split wait counters + NT cache hints.

typedef __attribute__((ext_vector_type(4))) float f32x4;
typedef int v4i __attribute__((vector_size(16)));
typedef __attribute__((address_space(1))) v4i* gv4i_p;  // global ptr to 16B
typedef __attribute__((address_space(3))) v4i* lv4i_p;  // LDS ptr to 16B

#define LIF_T 64
#define LIF_BN (32 * 16384) /* 524288 */
#define LIF_BLOCK 256
#define LIF_VEC 4
#define LIF_TILE (LIF_BLOCK * LIF_VEC) /* 1024 floats per block per step */
#define LIF_NBUF 4
#define LIF_PRE 3

#if __has_builtin(__builtin_amdgcn_global_load_async_to_lds_b128)
#define LIF_ASYNC 1
#else
#define LIF_ASYNC 0
#endif

// One LIF update on 4 lanes-worth of state. contract(off) is essential:
// HIP's default -ffp-contract=fast would fuse mul+add into v_fma_f32, which
// rounds differently from the reference's separate mul-then-add; a 1-ulp
// difference at the v_th compare flips a spike (error 1.0) and diverges that
// neuron's entire remaining trajectory.
__device__ __forceinline__ void lif_update(f32x4& v, const f32x4& cur,
                                           f32x4& s, float beta, float vth,
                                           float vrst) {
#pragma clang fp contract(off)
#pragma unroll
  for (int k = 0; k < 4; ++k) {
    const float t1 = v[k] * beta;
    const float vk = t1 + cur[k];
    const bool fire = vk > vth;
    s[k] = fire ? 1.0f : 0.0f;
    v[k] = fire ? vrst : vk;
  }
}

#if LIF_ASYNC

#if __has_builtin(__builtin_amdgcn_s_wait_asynccnt)
#define LIF_WAIT_ASYNC(n) __builtin_amdgcn_s_wait_asynccnt(n)
#else
#define LIF_WAIT_ASYNC(n) asm volatile("s_wait_asynccnt %0" ::"n"(n) : "memory")
#endif

// Deterministically close the WAR race: a pending ds_load of ring slot X must
// complete before the async DMA overwrites slot X.
#define LIF_WAIT_DS0() asm volatile("s_wait_dscnt 0" ::: "memory")

__device__ __forceinline__ void lif_issue_async(const float* g, float* l) {
  // Each lane DMAs its own 16B: global -> its private LDS slot. ASYNCcnt.
  __builtin_amdgcn_global_load_async_to_lds_b128(
      (gv4i_p)g, (lv4i_p)l, /*imm offset*/ 0, /*cpol*/ 0);
}

template <int t>
__device__ __forceinline__ void lif_step(const float* gsrc, float* spk,
                                         float (*lbuf)[LIF_TILE], int loff,
                                         f32x4& v, float beta, float vth,
                                         float vrst) {
  if constexpr (t < LIF_T) {
    if constexpr (t + LIF_PRE < LIF_T) {
      LIF_WAIT_DS0();
      lif_issue_async(gsrc + (size_t)(t + LIF_PRE) * LIF_BN,
                      &lbuf[(t + LIF_PRE) % LIF_NBUF][loff]);
    }
    // Async loads complete in order: allowing kOut outstanding guarantees
    // timestep t's tile has landed in LDS.
    constexpr int kOut =
        (LIF_T - 1 - t) < LIF_PRE ? (LIF_T - 1 - t) : LIF_PRE;
    LIF_WAIT_ASYNC(kOut);

    const f32x4 cur = *(const f32x4*)(&lbuf[t % LIF_NBUF][loff]);
    f32x4 s;
    lif_update(v, cur, s, beta, vth, vrst);
    __builtin_nontemporal_store(s, (f32x4*)(spk + (size_t)t * LIF_BN));
    lif_step<t + 1>(gsrc, spk, lbuf, loff, v, beta, vth, vrst);
  }
}

#endif  // LIF_ASYNC

__global__ __launch_bounds__(LIF_BLOCK) void lif_forward(
    const float* __restrict__ in, const float* __restrict__ p_beta,
    const float* __restrict__ p_vth, const float* __restrict__ p_vrst,
    float* __restrict__ out) {
  const float beta = *p_beta;
  const float vth = *p_vth;
  const float vrst = *p_vrst;

  const int loff = (int)threadIdx.x * LIF_VEC;
  const size_t base = (size_t)blockIdx.x * LIF_TILE + (size_t)loff;

  const float* gsrc = in + base;
  float* spk = out + base;                           // spikes [T,B,N]
  float* vfin = out + (size_t)LIF_T * LIF_BN + base; // v_final [B,N]

  f32x4 v = {0.0f, 0.0f, 0.0f, 0.0f};

#if LIF_ASYNC
  __shared__ float lbuf[LIF_NBUF][LIF_TILE];  // 16 KB ring, 4 timesteps deep
#pragma unroll
  for (int t = 0; t < LIF_PRE; ++t)
    lif_issue_async(gsrc + (size_t)t * LIF_BN, &lbuf[t][loff]);
  lif_step<0>(gsrc, spk, lbuf, loff, v, beta, vth, vrst);
#else
  // Fallback: register-double-buffered direct b128 loads + HW prefetch.
  const f32x4* gv = (const f32x4*)gsrc;
  const size_t stride4 = LIF_BN / 4;
  f32x4 nxt = __builtin_nontemporal_load(gv);
#pragma unroll
  for (int t = 0; t < LIF_T; ++t) {
    const f32x4 cur = nxt;
    if (t + 1 < LIF_T)
      nxt = __builtin_nontemporal_load(gv + (size_t)(t + 1) * stride4);
    if (t + 2 < LIF_T)
      __builtin_prefetch(gv + (size_t)(t + 2) * stride4, 0, 0);
    f32x4 s;
    lif_update(v, cur, s, beta, vth, vrst);
    __builtin_nontemporal_store(s, (f32x4*)(spk + (size_t)t * LIF_BN));
  }
#endif

  *(f32x4*)vfin = v;
}

extern "C" void kernel_launch(void* const* d_in, const int* in_sizes, int n_in,
                              void* d_out, int out_size, void* d_ws,
                              size_t ws_size, hipStream_t stream) {
  (void)in_sizes;
  (void)n_in;
  (void)out_size;
  (void)d_ws;
  (void)ws_size;
  const float* in = (const float*)d_in[0];
  const float* beta = (const float*)d_in[1];
  const float* vth = (const float*)d_in[2];
  const float* vrst = (const float*)d_in[3];
  // d_in[4] (k_superspike) is unused in the forward pass.
  dim3 grid(LIF_BN / LIF_TILE);  // 512 blocks
  dim3 block(LIF_BLOCK);         // 8 wave32s
  lif_forward<<<grid, block, 0, stream>>>(in, beta, vth, vrst, (float*)d_out);
}